// HypergraphMPConv_40587440947834
// MI455X (gfx1250) — compile-verified
//
#include <hip/hip_runtime.h>
#include <stdint.h>

// HypergraphMPConv on MI455X (gfx1250), wave32 + WMMA f32 16x16x4.
// Bandwidth-bound on the 1 GiB dense H: exactly 3 H passes
// (Dv pass, H^T@z pass with fused De, H@t pass). Deterministic reductions.
// Double-buffered GLOBAL_LOAD_ASYNC_TO_LDS_B128 pipeline (ASYNCcnt) feeds LDS;
// no VGPR staging -> no scratch spills; one barrier per K-chunk.

#define N_NODES 32768
#define M_EDGES 8192
#define IN_DIM  256
#define OUT_DIM 64

typedef float v2f __attribute__((ext_vector_type(2)));
typedef float v8f __attribute__((ext_vector_type(8)));

// Async DMA: copy 16 contiguous bytes global -> LDS (per lane), ASYNCcnt-tracked.
// VDST = 32-bit LDS byte address (generic shared pointer bits [31:0] per ISA 10.2).
__device__ __forceinline__ void async_b128(const float* gsrc, float* ldst) {
    uint32_t laddr = (uint32_t)(uintptr_t)ldst;
    asm volatile("global_load_async_to_lds_b128 %0, %1, off"
                 :: "v"(laddr), "v"(gsrc) : "memory");
}
__device__ __forceinline__ void wait_async0() {
    asm volatile("s_wait_asynccnt 0x0" ::: "memory");
}

// ---------------------------------------------------------------------------
// Pass 1: Dv[n] = sum_m H[n,m]*w[m];  dvis[n] = rsqrt(max(Dv,1e-6)).
// One 256-thread block per row; float4 coalesced reads; fixed-order reduce.
// ---------------------------------------------------------------------------
__global__ __launch_bounds__(256) void dv_kernel(const float* __restrict__ H,
                                                 const float* __restrict__ w,
                                                 float* __restrict__ dvis) {
    const int n = blockIdx.x;
    const int tid = threadIdx.x;
    const float4* __restrict__ Hrow = reinterpret_cast<const float4*>(H + (size_t)n * M_EDGES);
    const float4* __restrict__ w4   = reinterpret_cast<const float4*>(w);
    float s = 0.f;
#pragma unroll
    for (int j = 0; j < 8; ++j) {              // 8192/4 = 2048 = 256*8
        int i = tid + 256 * j;
        float4 h = Hrow[i];
        float4 ww = w4[i];
        s += h.x * ww.x + h.y * ww.y + h.z * ww.z + h.w * ww.w;
    }
    __shared__ float red[256];
    red[tid] = s;
    __syncthreads();
    for (int off = 128; off > 0; off >>= 1) {
        if (tid < off) red[tid] += red[tid + off];
        __syncthreads();
    }
    if (tid == 0) dvis[n] = rsqrtf(fmaxf(red[0], 1e-6f));
}

// ---------------------------------------------------------------------------
// Unified WMMA GEMM: C[row, o] = sum_k A(row,k) * B(k,o)   (o = 0..63)
// MODE 0: A(n,k)=x[n*256+k],  B=W_lin, out z = (acc + b[o]) * dvis[n]
// MODE 1: A(m,k)=H[k*8192+m], B=z,     out t = acc * (w[m]/max(De[m],1))
//         (De[m] read back from the LDS tile, fixed order -> deterministic)
// MODE 2: A(n,k)=H[n*8192+k], B=t,     out   = acc * dvis[n]
//
// Block: 256 threads = 8 waves; block tile = 32 rows x 64 cols, full-K loop.
// Wave w: rows 16*(w&1)..+15, cols 16*(w>>1)..+15 -> one v8f accumulator.
// LDS (double buffered, filled by async DMA):
//   MODE0/2 A: [32 r][64 k] pitch 68 -> A frag = 1 ds_load_b64, banks
//              4*lo+c..+3 cover all 64 banks once per wave (conflict-free).
//   MODE1   A: [64 k][32 m] pitch 40 -> 2x b32, half-groups +16 banks apart.
//   B:         [64 k][64 o] pitch 72 -> 2x b32, half-groups +16 banks apart.
//   All row pitches are multiples of 4 floats so async 16B chunks stay aligned.
// ---------------------------------------------------------------------------
template <int MODE>
__global__ __launch_bounds__(256) void wmma_gemm_kernel(const float* __restrict__ Asrc,
                                                        const float* __restrict__ Bsrc,
                                                        const float* __restrict__ aux,   // MODE0: b_lin, MODE1: w
                                                        const float* __restrict__ dvis,  // MODE0/2
                                                        float* __restrict__ Out,
                                                        int K, int aStride) {
    constexpr int APITCH = (MODE == 1) ? 40 : 68;
    constexpr int ASZ    = (MODE == 1) ? 64 * 40 : 32 * 68;
    constexpr int BPITCH = 72;
    constexpr int BSZ    = 64 * BPITCH;
    __shared__ float ldsA[2][ASZ];
    __shared__ float ldsB[2][BSZ];
    __shared__ float deLds[256];
    __shared__ float scLds[32];

    const int tid  = threadIdx.x;
    const int lane = tid & 31;
    const int wid  = tid >> 5;
    const int mb   = (wid & 1) << 4;   // row base within 32-row tile
    const int ob   = (wid >> 1) << 4;  // col base within 64 cols
    const int hi   = lane >> 4;        // lane group: K pair select
    const int lo   = lane & 15;
    const int row0 = blockIdx.x * 32;

    v8f acc = {};
    float deAcc = 0.f;

    // ---- async issue of one 64-deep K chunk into buffer `b` ----
    auto issue_chunk = [&](int kc, int b) {
        if (MODE == 1) {
            // A(m,k)=H[k*aStride+m]: 64 k-rows x 128B; 8 lanes/row x b128.
#pragma unroll
            for (int j = 0; j < 2; ++j) {
                int kk = (tid >> 3) + 32 * j;     // 0..63
                int ms = (tid & 7) << 2;          // m segment (x4 floats)
                async_b128(Asrc + (size_t)(kc + kk) * aStride + row0 + ms,
                           &ldsA[b][kk * APITCH + ms]);
            }
        } else {
            // A(r,k): 32 rows x 256B; 16 lanes/row x b128.
#pragma unroll
            for (int j = 0; j < 2; ++j) {
                int idx = tid + 256 * j;
                int rr = idx >> 4;                // 0..31
                int ks = (idx & 15) << 2;         // k segment (x4 floats)
                async_b128(Asrc + (size_t)(row0 + rr) * aStride + kc + ks,
                           &ldsA[b][rr * APITCH + ks]);
            }
        }
        // B(k,o): 64 rows x 256B; 16 lanes/row x b128.
#pragma unroll
        for (int j = 0; j < 4; ++j) {
            int idx = tid + 256 * j;
            int kk = idx >> 4;                    // 0..63
            int os = (idx & 15) << 2;             // o segment (x4 floats)
            async_b128(Bsrc + (size_t)(kc + kk) * 64 + os,
                       &ldsB[b][kk * BPITCH + os]);
        }
    };

    issue_chunk(0, 0);

    int buf = 0;
    for (int kc = 0; kc < K; kc += 64) {
        wait_async0();        // my async writes into ldsX[buf] are complete
        __syncthreads();      // everyone's are; prior reads of buf^1 are done

        const int kn = kc + 64;
        if (kn < K) {
            issue_chunk(kn, buf ^ 1);   // DMA next chunk while we compute
            // look-ahead prefetch of the H stream, 2 chunks out
            if (MODE != 0 && kc + 128 < K && tid < 64) {
                if (MODE == 1)
                    __builtin_prefetch(Asrc + (size_t)(kc + 128 + tid) * aStride + row0, 0, 0);
                else
                    __builtin_prefetch(Asrc + (size_t)(row0 + (tid & 31)) * aStride + kc + 128 + ((tid >> 5) << 5), 0, 0);
            }
        }

        if (MODE == 1) {
            // fused De: fixed-order re-read of this chunk's A tile (H values)
#pragma unroll
            for (int j = 0; j < 8; ++j)
                deAcc += ldsA[buf][((tid >> 5) + 8 * j) * APITCH + (tid & 31)];
        }

        // ---- 16 WMMA steps over the 64-deep K chunk ----
#pragma unroll
        for (int ks = 0; ks < 64; ks += 4) {
            const int ka = ks + 2 * hi;  // lanes 0-15: K {ks,ks+1}; 16-31: {ks+2,ks+3}
            v2f a, b;
            if (MODE == 1) {
                a.x = ldsA[buf][(ka + 0) * APITCH + mb + lo];
                a.y = ldsA[buf][(ka + 1) * APITCH + mb + lo];
            } else {
                a = *reinterpret_cast<const v2f*>(&ldsA[buf][(mb + lo) * APITCH + ka]);
            }
            b.x = ldsB[buf][(ka + 0) * BPITCH + ob + lo];
            b.y = ldsB[buf][(ka + 1) * BPITCH + ob + lo];
            acc = __builtin_amdgcn_wmma_f32_16x16x4_f32(
                /*neg_a=*/false, a, /*neg_b=*/false, b,
                /*c_mod=*/(short)0, acc, /*reuse_a=*/false, /*reuse_b=*/false);
        }
        buf ^= 1;
    }

    // ---- epilogues ----
    if (MODE == 1) {
        __syncthreads();
        // Deterministic De reduce across the 8 thread-groups per column.
        deLds[tid] = deAcc;
        __syncthreads();
        if (tid < 32) {
            float s = 0.f;
#pragma unroll
            for (int g = 0; g < 8; ++g) s += deLds[g * 32 + tid];
            s = fmaxf(s, 1.0f);                 // clip(De, 1, inf)
            scLds[tid] = aux[row0 + tid] / s;   // w[m] / De[m]
        }
        __syncthreads();
#pragma unroll
        for (int v = 0; v < 8; ++v) {
            int m = mb + v + 8 * hi;            // D layout: vgpr v -> row v or v+8
            Out[(size_t)(row0 + m) * 64 + ob + lo] = acc[v] * scLds[m];
        }
    } else if (MODE == 0) {
        float bb = aux[ob + lo];
#pragma unroll
        for (int v = 0; v < 8; ++v) {
            int r = row0 + mb + v + 8 * hi;
            Out[(size_t)r * 64 + ob + lo] = (acc[v] + bb) * dvis[r];
        }
    } else {
#pragma unroll
        for (int v = 0; v < 8; ++v) {
            int r = row0 + mb + v + 8 * hi;
            Out[(size_t)r * 64 + ob + lo] = acc[v] * dvis[r];
        }
    }
}

// ---------------------------------------------------------------------------
extern "C" void kernel_launch(void* const* d_in, const int* in_sizes, int n_in,
                              void* d_out, int out_size, void* d_ws, size_t ws_size,
                              hipStream_t stream) {
    const float* x     = (const float*)d_in[0];  // [N, 256]
    const float* H     = (const float*)d_in[1];  // [N, M]
    const float* w     = (const float*)d_in[2];  // [M]
    const float* W_lin = (const float*)d_in[3];  // [256, 64]
    const float* b_lin = (const float*)d_in[4];  // [64]
    float* out = (float*)d_out;                  // [N, 64]

    float* ws   = (float*)d_ws;
    float* dvis = ws;                                   // N floats
    float* z    = dvis + N_NODES;                       // N*64 floats
    float* t    = z + (size_t)N_NODES * OUT_DIM;        // M*64 floats

    // Pass 1: weighted node degrees -> dv^{-1/2}          (1 GiB H read)
    dv_kernel<<<N_NODES, 256, 0, stream>>>(H, w, dvis);

    // z = dvis * (x @ W_lin + b)                          (32 MB x read)
    wmma_gemm_kernel<0><<<N_NODES / 32, 256, 0, stream>>>(
        x, W_lin, b_lin, dvis, z, IN_DIM, IN_DIM);

    // t = (w/De) * (H^T @ z), De fused                    (1 GiB H read)
    wmma_gemm_kernel<1><<<M_EDGES / 32, 256, 0, stream>>>(
        H, z, w, nullptr, t, N_NODES, M_EDGES);

    // out = dvis * (H @ t)                                (1 GiB H read)
    wmma_gemm_kernel<2><<<N_NODES / 32, 256, 0, stream>>>(
        H, t, nullptr, dvis, out, M_EDGES, M_EDGES);
}